// SimpleMambaBlock_74766790689538
// MI455X (gfx1250) — compile-verified
//
#include <hip/hip_runtime.h>
#include <hip/hip_bf16.h>
#include <math.h>

// ---------------- Problem dims ----------------
#define D_MODEL  1024
#define D_INNER  2048
#define D_STATE  16
#define D_CONV   4
#define DT_RANK  64
#define BATCH    2
#define SEQ      2048
#define ROWS     (BATCH * SEQ)        // 4096
#define LN_EPS   1e-5f

typedef __attribute__((ext_vector_type(2))) float v2f;
typedef __attribute__((ext_vector_type(4))) float v4f;
typedef __attribute__((ext_vector_type(8))) float v8f;

// =====================================================================
// fp32 WMMA GEMM:  Y[M x N] = A[M x K] * W[N x K]^T
// One wave computes a 32x64 tile (2 M-subtiles x 4 N-subtiles,
// 8 accumulators): per k-step 2 A + 4 B fragment loads feed 8
// v_wmma_f32_16x16x4_f32 ops -> 0.75 fragment loads per WMMA.
//   epilogue: 0 = plain store; 1 = softplus(v + bias[col])
// =====================================================================
__global__ __launch_bounds__(256)
void gemm_f32_wmma(const float* __restrict__ A, const float* __restrict__ W,
                   float* __restrict__ Y,
                   int M, int N, int K, int lda, int ldw, int ldy,
                   const float* __restrict__ bias, int epilogue)
{
    const int wave = threadIdx.x >> 5;
    const int lane = threadIdx.x & 31;
    const int half = lane >> 4;       // 0: lanes 0-15, 1: lanes 16-31
    const int l16  = lane & 15;

    const int tg = blockIdx.x * (blockDim.x >> 5) + wave;  // group of 4 N-tiles
    const int tm = blockIdx.y;                             // 32-row M tile
    const int n0 = tg << 6;                                // 64 cols per wave
    const int m0 = tm << 5;                                // 32 rows per wave
    if (n0 >= N) return;   // wave-uniform: EXEC stays all-ones for WMMA

    // A fragments: lane l16 holds row; VGPR j holds K = k + 2*half + j
    const float* __restrict__ ar0 = A + (size_t)(m0 + l16) * lda + 2 * half;
    const float* __restrict__ ar1 = A + (size_t)(m0 + 16 + l16) * lda + 2 * half;

    // B fragments from W^T; clamp invalid sub-tiles to row 0 (computed, not stored)
    bool valid[4];
    int nb0 = n0, nb1 = n0 + 16, nb2 = n0 + 32, nb3 = n0 + 48;
    valid[0] = nb0 < N; valid[1] = nb1 < N; valid[2] = nb2 < N; valid[3] = nb3 < N;
    const float* __restrict__ wr0 = W + (size_t)((valid[0] ? nb0 : 0) + l16) * ldw + 2 * half;
    const float* __restrict__ wr1 = W + (size_t)((valid[1] ? nb1 : 0) + l16) * ldw + 2 * half;
    const float* __restrict__ wr2 = W + (size_t)((valid[2] ? nb2 : 0) + l16) * ldw + 2 * half;
    const float* __restrict__ wr3 = W + (size_t)((valid[3] ? nb3 : 0) + l16) * ldw + 2 * half;

    v8f acc0[4] = {};   // rows m0..m0+15
    v8f acc1[4] = {};   // rows m0+16..m0+31
#pragma unroll 2
    for (int k = 0; k < K; k += 4) {
        v2f a0 = *(const v2f*)(ar0 + k);
        v2f a1 = *(const v2f*)(ar1 + k);
        v2f b0 = *(const v2f*)(wr0 + k);
        v2f b1 = *(const v2f*)(wr1 + k);
        v2f b2 = *(const v2f*)(wr2 + k);
        v2f b3 = *(const v2f*)(wr3 + k);
        acc0[0] = __builtin_amdgcn_wmma_f32_16x16x4_f32(false, a0, false, b0, (short)0, acc0[0], false, false);
        acc1[0] = __builtin_amdgcn_wmma_f32_16x16x4_f32(false, a1, false, b0, (short)0, acc1[0], false, false);
        acc0[1] = __builtin_amdgcn_wmma_f32_16x16x4_f32(false, a0, false, b1, (short)0, acc0[1], false, false);
        acc1[1] = __builtin_amdgcn_wmma_f32_16x16x4_f32(false, a1, false, b1, (short)0, acc1[1], false, false);
        acc0[2] = __builtin_amdgcn_wmma_f32_16x16x4_f32(false, a0, false, b2, (short)0, acc0[2], false, false);
        acc1[2] = __builtin_amdgcn_wmma_f32_16x16x4_f32(false, a1, false, b2, (short)0, acc1[2], false, false);
        acc0[3] = __builtin_amdgcn_wmma_f32_16x16x4_f32(false, a0, false, b3, (short)0, acc0[3], false, false);
        acc1[3] = __builtin_amdgcn_wmma_f32_16x16x4_f32(false, a1, false, b3, (short)0, acc1[3], false, false);
    }

    // C/D layout: VGPR r, lanes 0-15 -> M = base+r, lanes 16-31 -> M = base+r+8
#pragma unroll
    for (int i = 0; i < 4; ++i) {
        if (!valid[i]) continue;          // wave-uniform
        const int col = n0 + (i << 4) + l16;
        const float bcol = (epilogue == 1) ? bias[col] : 0.0f;
#pragma unroll
        for (int r = 0; r < 8; ++r) {
            int row0 = m0 + r + (half << 3);
            float v0 = acc0[i][r];
            float v1 = acc1[i][r];
            if (epilogue == 1) {
                v0 += bcol;
                v0 = (v0 > 20.0f) ? v0 : log1pf(__expf(v0));   // softplus
                v1 += bcol;
                v1 = (v1 > 20.0f) ? v1 : log1pf(__expf(v1));
            }
            Y[(size_t)row0 * ldy + col]        = v0;
            Y[(size_t)(row0 + 16) * ldy + col] = v1;
        }
    }
}

// =====================================================================
// Depthwise causal conv (width 4) + SiLU.
//   xz : [ROWS][2*D_INNER] (xi = cols 0..D_INNER-1)
//   xc : [ROWS][D_INNER]
// =====================================================================
__global__ __launch_bounds__(256)
void conv_silu_kernel(const float* __restrict__ xz,
                      const float* __restrict__ cw,
                      const float* __restrict__ cb,
                      float* __restrict__ xc)
{
    int idx = blockIdx.x * blockDim.x + threadIdx.x;
    if (idx >= ROWS * D_INNER) return;
    int d   = idx & (D_INNER - 1);
    int row = idx >> 11;             // b*SEQ + l
    int l   = row & (SEQ - 1);

    float acc = cb[d];
    const float* __restrict__ w = cw + d * D_CONV;
#pragma unroll
    for (int j = 0; j < D_CONV; ++j) {
        int ls = l - (D_CONV - 1) + j;
        if (ls >= 0)
            acc = fmaf(w[j], xz[(size_t)(row - (D_CONV - 1) + j) * (2 * D_INNER) + d], acc);
    }
    xc[idx] = acc / (1.0f + __expf(-acc));   // silu
}

// =====================================================================
// Selective scan. One thread per (b, d). h[16] lives in registers.
// B_t / C_t are loaded as 8x float4 (16B-aligned: 64*4 within 384B rows).
//   dt_y: in = dt, out = y = (scan_y + xc*D) * silu(z)   (in-place)
// =====================================================================
__global__ __launch_bounds__(256)
void scan_kernel(const float* __restrict__ xz,
                 const float* __restrict__ xc,
                 const float* __restrict__ xdbl,
                 float* __restrict__ dt_y,
                 const float* __restrict__ A_log,
                 const float* __restrict__ Dv)
{
    int t = blockIdx.x * blockDim.x + threadIdx.x;
    if (t >= BATCH * D_INNER) return;
    int b = t >> 11;
    int d = t & (D_INNER - 1);

    float Aneg[D_STATE];
#pragma unroll
    for (int s = 0; s < D_STATE; ++s)
        Aneg[s] = -__expf(A_log[d * D_STATE + s]);
    const float Dd = Dv[d];

    float h[D_STATE];
#pragma unroll
    for (int s = 0; s < D_STATE; ++s) h[s] = 0.0f;

    const size_t rbase = (size_t)b * SEQ;
    for (int l = 0; l < SEQ; ++l) {
        const size_t row = rbase + l;
        const float dtv = dt_y[row * D_INNER + d];
        const float xv  = xc[row * D_INNER + d];

        const v4f* __restrict__ bc =
            (const v4f*)(xdbl + row * (DT_RANK + 2 * D_STATE) + DT_RANK);
        v4f Bq[4], Cq[4];
#pragma unroll
        for (int q = 0; q < 4; ++q) { Bq[q] = bc[q]; Cq[q] = bc[4 + q]; }

        const float dx = dtv * xv;
        float y = 0.0f;
#pragma unroll
        for (int s = 0; s < D_STATE; ++s) {
            float Bs = Bq[s >> 2][s & 3];
            float Cs = Cq[s >> 2][s & 3];
            float dA = __expf(dtv * Aneg[s]);
            h[s] = fmaf(dA, h[s], dx * Bs);
            y = fmaf(h[s], Cs, y);
        }
        float zv = xz[row * (2 * D_INNER) + D_INNER + d];
        y = (y + xv * Dd) * (zv / (1.0f + __expf(-zv)));
        dt_y[row * D_INNER + d] = y;
    }
}

// =====================================================================
// Residual + LayerNorm: one block (256 thr) per row of D_MODEL=1024.
// =====================================================================
__global__ __launch_bounds__(256)
void ln_kernel(const float* __restrict__ outw,
               const float* __restrict__ x,
               const float* __restrict__ gamma,
               const float* __restrict__ beta,
               float* __restrict__ out)
{
    const int row = blockIdx.x;
    const int tid = threadIdx.x;
    __shared__ float s1[256], s2[256];

    float v[4];
    float sum = 0.0f, sumsq = 0.0f;
#pragma unroll
    for (int i = 0; i < 4; ++i) {
        int c = tid + i * 256;
        float h = outw[(size_t)row * D_MODEL + c] + x[(size_t)row * D_MODEL + c];
        v[i] = h;
        sum += h;
        sumsq = fmaf(h, h, sumsq);
    }
    s1[tid] = sum; s2[tid] = sumsq;
    __syncthreads();
    for (int off = 128; off > 0; off >>= 1) {
        if (tid < off) { s1[tid] += s1[tid + off]; s2[tid] += s2[tid + off]; }
        __syncthreads();
    }
    const float mu   = s1[0] * (1.0f / D_MODEL);
    const float var  = s2[0] * (1.0f / D_MODEL) - mu * mu;
    const float rstd = rsqrtf(var + LN_EPS);
#pragma unroll
    for (int i = 0; i < 4; ++i) {
        int c = tid + i * 256;
        out[(size_t)row * D_MODEL + c] = (v[i] - mu) * rstd * gamma[c] + beta[c];
    }
}

// =====================================================================
extern "C" void kernel_launch(void* const* d_in, const int* in_sizes, int n_in,
                              void* d_out, int out_size, void* d_ws, size_t ws_size,
                              hipStream_t stream)
{
    const float* x         = (const float*)d_in[0];   // [2,2048,1024]
    const float* in_proj_w = (const float*)d_in[1];   // [4096,1024]
    const float* conv_w    = (const float*)d_in[2];   // [2048,1,4]
    const float* conv_b    = (const float*)d_in[3];   // [2048]
    const float* x_proj_w  = (const float*)d_in[4];   // [96,2048]
    const float* dt_proj_w = (const float*)d_in[5];   // [2048,64]
    const float* dt_proj_b = (const float*)d_in[6];   // [2048]
    const float* A_log     = (const float*)d_in[7];   // [2048,16]
    const float* Dv        = (const float*)d_in[8];   // [2048]
    const float* out_proj_w= (const float*)d_in[9];   // [1024,2048]
    const float* ln_gamma  = (const float*)d_in[10];  // [1024]
    const float* ln_beta   = (const float*)d_in[11];  // [1024]
    float* out = (float*)d_out;                       // [4096,1024]

    // Workspace partition (floats)
    float* ws   = (float*)d_ws;
    float* xz   = ws;                         // 4096*4096 = 16,777,216
    float* xc   = ws + 16777216;              // 4096*2048 =  8,388,608
    float* xdbl = ws + 25165824;              // 4096*96   =    393,216
    float* dty  = ws + 25559040;              // 4096*2048 =  8,388,608 (dt then y)
    float* outw = ws + 33947648;              // 4096*1024 =  4,194,304

    const dim3 blk(256);
    const int wavesPerBlk = 8;                // 256 / 32
    const int mtiles = ROWS / 32;             // 128

    // 1) in_proj: xz[4096,4096] = x[4096,1024] @ in_proj_w[4096,1024]^T
    {
        int ngrp = (2 * D_INNER + 63) / 64;   // 64
        gemm_f32_wmma<<<dim3((ngrp + wavesPerBlk - 1) / wavesPerBlk, mtiles), blk, 0, stream>>>(
            x, in_proj_w, xz, ROWS, 2 * D_INNER, D_MODEL,
            D_MODEL, D_MODEL, 2 * D_INNER, nullptr, 0);
    }

    // 2) causal depthwise conv + silu -> xc[4096,2048]
    conv_silu_kernel<<<(ROWS * D_INNER + 255) / 256, blk, 0, stream>>>(
        xz, conv_w, conv_b, xc);

    // 3) x_proj: xdbl[4096,96] = xc[4096,2048] @ x_proj_w[96,2048]^T
    {
        int ngrp = (DT_RANK + 2 * D_STATE + 63) / 64;   // 2
        gemm_f32_wmma<<<dim3((ngrp + wavesPerBlk - 1) / wavesPerBlk, mtiles), blk, 0, stream>>>(
            xc, x_proj_w, xdbl, ROWS, DT_RANK + 2 * D_STATE, D_INNER,
            D_INNER, D_INNER, DT_RANK + 2 * D_STATE, nullptr, 0);
    }

    // 4) dt: dty[4096,2048] = softplus(xdbl[:, :64] @ dt_proj_w[2048,64]^T + b)
    {
        int ngrp = (D_INNER + 63) / 64;       // 32
        gemm_f32_wmma<<<dim3((ngrp + wavesPerBlk - 1) / wavesPerBlk, mtiles), blk, 0, stream>>>(
            xdbl, dt_proj_w, dty, ROWS, D_INNER, DT_RANK,
            DT_RANK + 2 * D_STATE, DT_RANK, D_INNER, dt_proj_b, 1);
    }

    // 5) selective scan + gating -> y written in-place into dty
    scan_kernel<<<(BATCH * D_INNER + 255) / 256, blk, 0, stream>>>(
        xz, xc, xdbl, dty, A_log, Dv);

    // 6) out_proj: outw[4096,1024] = y[4096,2048] @ out_proj_w[1024,2048]^T
    {
        int ngrp = (D_MODEL + 63) / 64;       // 16
        gemm_f32_wmma<<<dim3((ngrp + wavesPerBlk - 1) / wavesPerBlk, mtiles), blk, 0, stream>>>(
            dty, out_proj_w, outw, ROWS, D_MODEL, D_INNER,
            D_INNER, D_INNER, D_MODEL, nullptr, 0);
    }

    // 7) residual + layernorm -> out
    ln_kernel<<<ROWS, blk, 0, stream>>>(outw, x, ln_gamma, ln_beta, out);
}